// GrCNetConvOnly_7962869367672
// MI455X (gfx1250) — compile-verified
//
#include <hip/hip_runtime.h>

// CDNA5 / gfx1250, wave32.
typedef __attribute__((ext_vector_type(2))) float v2f;
typedef __attribute__((ext_vector_type(8))) float v8f;

#define DIM   400   // embedding dim
#define NC    50    // conv output channels
#define NDT   25    // DIM / 16 d-tiles
#define NCT   4     // ceil(NC / 16) c-tiles

__global__ __launch_bounds__(256)
void grcnet_fused_kernel(const float* __restrict__ entity_emb,
                         const float* __restrict__ relation_emb,
                         const float* __restrict__ conv_w,   // (NC,1,1,3) -> [c*3 + k]
                         const float* __restrict__ conv_b,   // (NC,)
                         const float* __restrict__ fc_w,     // (1, NC*DIM) -> [c*DIM + d]
                         const float* __restrict__ fc_b,     // (1,)
                         const int*   __restrict__ batch_inputs, // (B,3) h,r,t
                         float*       __restrict__ out)      // (B,)
{
    const int b    = blockIdx.x;
    const int tid  = threadIdx.x;
    const int wave = tid >> 5;        // 0..7
    const int lane = tid & 31;
    const int m    = lane & 15;       // row/col within 16
    const bool hi  = lane >= 16;

    const int head = batch_inputs[b * 3 + 0];
    const int rel  = batch_inputs[b * 3 + 1];
    const int tail = batch_inputs[b * 3 + 2];

    const float* __restrict__ hrow = entity_emb   + (long)head * DIM;
    const float* __restrict__ rrow = relation_emb + (long)rel  * DIM;
    const float* __restrict__ trow = entity_emb   + (long)tail * DIM;

    // Branchless per-lane A-row source: low lanes read h (K=0), high lanes t (K=2).
    const float* __restrict__ arow = hi ? trow : hrow;

    // ---- B tiles: W^T (K=4 x N=16) per c-tile; invariant over d ----
    // 32-bit B layout mirrors A: lanes 0-15 hold K=0 (vgpr0), K=1 (vgpr1);
    // lanes 16-31 hold K=2 (vgpr0), K=3 (vgpr1). K order = {h,r,t,pad}.
    v2f Bt[NCT];
    v8f bias[NCT];
    #pragma unroll
    for (int ct = 0; ct < NCT; ++ct) {
        const int c = ct * 16 + m;
        float b0 = 0.0f, b1 = 0.0f, bb = 0.0f;
        if (c < NC) {
            b0 = hi ? conv_w[c * 3 + 2] : conv_w[c * 3 + 0];
            b1 = hi ? 0.0f              : conv_w[c * 3 + 1];
            bb = conv_b[c];
        }
        Bt[ct].x = b0;
        Bt[ct].y = b1;
        v8f bv;
        #pragma unroll
        for (int r = 0; r < 8; ++r) bv[r] = bb;   // C input = channel bias
        bias[ct] = bv;
    }

    // ---- loop over this wave's d-tiles ----
    float partial = 0.0f;
    for (int dt = wave; dt < NDT; dt += 8) {      // wave-uniform trip count
        const int d0 = dt * 16;

        // A (M=16 x K=4): lanes 0-15: (h[d], r[d]); lanes 16-31: (t[d], 0).
        // Both loads unconditional; high lanes' K=3 slot zeroed via select.
        v2f A;
        A.x = arow[d0 + m];
        float a1 = rrow[d0 + m];
        A.y = hi ? 0.0f : a1;                     // v_cndmask, no exec branch

        const int drow0 = d0 + (hi ? 8 : 0);      // rows this lane holds in D

        #pragma unroll
        for (int ct = 0; ct < NCT; ++ct) {
            // D = A x B + bias  (16x16 f32 accumulator, 8 VGPRs)
            v8f acc = __builtin_amdgcn_wmma_f32_16x16x4_f32(
                /*neg_a=*/false, A, /*neg_b=*/false, Bt[ct],
                /*c_mod=*/(short)0, bias[ct],
                /*reuse_a=*/false, /*reuse_b=*/false);

            const int c = ct * 16 + m;            // this lane's channel (N)
            if (c < NC) {
                // fc_w[c*DIM + d] for 8 consecutive d: two 128-bit loads
                const float4* fwv =
                    reinterpret_cast<const float4*>(fc_w + c * DIM + drow0);
                const float4 f0 = fwv[0];
                const float4 f1 = fwv[1];
                const float fw[8] = { f0.x, f0.y, f0.z, f0.w,
                                      f1.x, f1.y, f1.z, f1.w };
                #pragma unroll
                for (int r = 0; r < 8; ++r) {
                    // ReLU as a single v_med3_f32: median(v, 0, +inf) = clamp(v, 0, inf)
                    const float v = __builtin_amdgcn_fmed3f(acc[r], 0.0f,
                                                            __builtin_inff());
                    partial = fmaf(v, fw[r], partial);
                }
            }
        }
    }

    // ---- wave32 reduction ----
    #pragma unroll
    for (int off = 16; off > 0; off >>= 1)
        partial += __shfl_xor(partial, off, 32);

    __shared__ float red[8];
    if (lane == 0) red[wave] = partial;
    __syncthreads();
    if (tid == 0) {
        float tot = fc_b[0];
        #pragma unroll
        for (int w = 0; w < 8; ++w) tot += red[w];
        out[b] = tot;
    }
}

extern "C" void kernel_launch(void* const* d_in, const int* in_sizes, int n_in,
                              void* d_out, int out_size, void* d_ws, size_t ws_size,
                              hipStream_t stream) {
    const float* entity_emb   = (const float*)d_in[0];
    const float* relation_emb = (const float*)d_in[1];
    const float* conv_w       = (const float*)d_in[2];
    const float* conv_b       = (const float*)d_in[3];
    const float* fc_w         = (const float*)d_in[4];
    const float* fc_b         = (const float*)d_in[5];
    const int*   batch_inputs = (const int*)d_in[6];
    float*       out          = (float*)d_out;

    const int B = out_size;   // 16384
    grcnet_fused_kernel<<<B, 256, 0, stream>>>(entity_emb, relation_emb,
                                               conv_w, conv_b, fc_w, fc_b,
                                               batch_inputs, out);
}